// DeformConv2D_42374147342739
// MI455X (gfx1250) — compile-verified
//
#include <hip/hip_runtime.h>
#include <math.h>

typedef __attribute__((ext_vector_type(2))) float v2f;
typedef __attribute__((ext_vector_type(8))) float v8f;

#define BB   8
#define CCH  64
#define HH   64
#define WW   64
#define OO   128
#define NN   9
#define KKT  576      // k = n*64 + c
#define HP   66       // padded H/W
#define KCH  192      // K chunk (3 kernel taps * 64 channels)
#define LDSS 194      // KCH + 2 pad (bank-conflict avoidance)
#define OSTR 68       // epilogue tile row stride (64 + 4 pad, keeps float4 alignment)

// ---------------- prep: NCHW -> padded NHWC ----------------
__global__ __launch_bounds__(256) void pad_nhwc_kernel(const float* __restrict__ x,
                                                       float* __restrict__ xp) {
    int t = blockIdx.x * 256 + threadIdx.x;
    if (t >= BB * HP * HP * CCH) return;
    int c  = t & 63;
    int r  = t >> 6;          // b*HP*HP + i*HP + j
    int j  = r % HP;
    int r2 = r / HP;
    int i  = r2 % HP;
    int b  = r2 / HP;
    float v = 0.f;
    if (i >= 1 && i <= HH && j >= 1 && j <= WW)
        v = x[(((size_t)b * CCH + c) * HH + (i - 1)) * WW + (j - 1)];
    xp[t] = v;
}

// -------- prep: conv_w (O,C,3,3) -> row-paired Wt2[(k/2)][o][k&1], k = n*64+c --------
// B fragment (rows k, k+1 at column o) becomes ONE 8-byte coalesced load.
__global__ __launch_bounds__(256) void wtr_kernel(const float* __restrict__ cw,
                                                  float* __restrict__ wt2) {
    int t = blockIdx.x * 256 + threadIdx.x;   // t == destination index
    if (t >= KKT * OO) return;
    int kl    = t & 1;
    int o     = (t >> 1) & 127;
    int kpair = t >> 8;
    int k     = kpair * 2 + kl;
    int n     = k >> 6;
    int c     = k & 63;
    wt2[t] = cw[((size_t)o * CCH + c) * NN + n];
}

// ---------------- offset + modulation convs (27 out channels) ----------------
__global__ __launch_bounds__(256) void offmod_kernel(const float* __restrict__ xp,
                                                     const float* __restrict__ shift_w,
                                                     const float* __restrict__ shift_b,
                                                     const float* __restrict__ mod_w,
                                                     const float* __restrict__ mod_b,
                                                     float* __restrict__ pm) {
    __shared__ float wl[27 * 577];          // padded stride 577: bank-conflict-free
    int tid = threadIdx.x;
    for (int t = tid; t < 27 * 576; t += 256) {
        int oc = t / 576;
        int r  = t - oc * 576;              // r = n*64 + c
        int n  = r >> 6;
        int c  = r & 63;
        float v = (oc < 18) ? shift_w[((size_t)oc * CCH + c) * NN + n]
                            : mod_w[((size_t)(oc - 18) * CCH + c) * NN + n];
        wl[oc * 577 + r] = v;
    }
    __syncthreads();

    int p   = blockIdx.x * 8 + (tid >> 5);  // pixel index  b*4096 + h*64 + w
    int oc  = tid & 31;                     // lanes 27..31 idle (clamped)
    int oce = (oc < 27) ? oc : 0;
    int b = p >> 12;
    int h = (p >> 6) & 63;
    int w = p & 63;

    float acc = 0.f;
    if (oc < 27) acc = (oc < 18) ? shift_b[oc] : mod_b[oc - 18];

    for (int n = 0; n < NN; ++n) {
        int qx = h + (n / 3);               // padded coords
        int qy = w + (n % 3);
        const float* xrow = xp + (((size_t)b * HP + qx) * HP + qy) * CCH;
        const float* wrow = &wl[oce * 577 + n * 64];
        for (int c = 0; c < CCH; ++c)
            acc = fmaf(wrow[c], xrow[c], acc);
    }
    if (oc < 27) {
        float outv = (oc < 18) ? acc : (1.f / (1.f + __expf(-acc)));
        pm[(size_t)p * 27 + oc] = outv;
    }
}

// ---------------- fused gather + WMMA f32 GEMM ----------------
// block = 256 threads (8 waves), owns one (b,h) row: 64 pixels x 128 outputs
__global__ __launch_bounds__(256) void deform_gemm_kernel(const float* __restrict__ xp,
                                                          const float* __restrict__ wt2,
                                                          const float* __restrict__ pm,
                                                          float* __restrict__ out) {
    __shared__ float smem[64 * LDSS];       // 49,664 B; reused as 128x68 out-tile

    const int tid   = threadIdx.x;
    const int b     = blockIdx.x >> 6;
    const int h     = blockIdx.x & 63;
    const int lane  = tid & 31;
    const int wv    = tid >> 5;
    const int msub  = wv & 3;               // which 16-pixel group
    const int ohalf = wv >> 2;              // which half of O
    const int row   = lane & 15;
    const int khi   = lane >> 4;            // lanes 16-31 hold K+2 / M+8 halves

    v8f acc[4] = {};

    const int grp = tid >> 6;               // 4 groups of 64 threads
    const int c   = tid & 63;
    const float* pmrow = pm + ((size_t)(b * 64 + h) * 64) * 27;
    const float* xb    = xp + (size_t)b * HP * HP * CCH;

    for (int nc = 0; nc < 3; ++nc) {
        // ---- gather chunk: taps n = nc*3 .. nc*3+2, 16 px per group, all 64 c ----
        for (int px = grp * 16; px < grp * 16 + 16; ++px) {
#pragma unroll
            for (int nl = 0; nl < 3; ++nl) {
                int n = nc * 3 + nl;
                float ox = pmrow[px * 27 + n];
                float oy = pmrow[px * 27 + n + 9];
                float mv = pmrow[px * 27 + 18 + n];
                float posx = ox + (float)(n / 3 - 1) + (float)(h + 1);
                float posy = oy + (float)(n % 3 - 1) + (float)(px + 1);
                float fx = floorf(posx), fy = floorf(posy);
                int ltx = (int)fminf(fmaxf(fx, 0.f), 63.f);
                int lty = (int)fminf(fmaxf(fy, 0.f), 63.f);
                int rbx = (int)fminf(fmaxf(fx + 1.f, 0.f), 63.f);
                int rby = (int)fminf(fmaxf(fy + 1.f, 0.f), 63.f);
                int dx = rbx - ltx, dy = rby - lty;
                // reference's integer-cast weights: g_lt = 1
                float g_rb = (float)((1 - dx) * (1 - dy));
                float g_lb = (float)(1 + dy);
                float g_rt = (float)(1 - dx);
                float vlt = xb[((size_t)ltx * HP + lty) * CCH + c];
                float vrb = xb[((size_t)rbx * HP + rby) * CCH + c];
                float vlb = xb[((size_t)ltx * HP + rby) * CCH + c];
                float vrt = xb[((size_t)rbx * HP + lty) * CCH + c];
                float val = (vlt + g_rb * vrb + g_lb * vlb + g_rt * vrt) * mv;
                smem[px * LDSS + nl * 64 + c] = val;
            }
        }
        __syncthreads();

        // ---- GEMM over this K chunk (paired-row Wt2: one b64 per B fragment) ----
        const float* wchunk = wt2 + (size_t)(nc * KCH) * OO;   // == kbase*128 base
        const float* arow   = &smem[(msub * 16 + row) * LDSS + 2 * khi];
        for (int kk = 0; kk < KCH; kk += 4) {
            v2f a;
            a.x = arow[kk];
            a.y = arow[kk + 1];
            int kbase = kk + 2 * khi;                          // even
#pragma unroll
            for (int j = 0; j < 4; ++j) {
                int o = ohalf * 64 + j * 16 + row;
                v2f bf = *(const v2f*)(wchunk + (size_t)kbase * 128 + o * 2);
                acc[j] = __builtin_amdgcn_wmma_f32_16x16x4_f32(
                    false, a, false, bf, (short)0, acc[j], false, false);
            }
        }
        __syncthreads();
    }

    // ---- epilogue: transpose through LDS, then coalesced float4 stores ----
    float* Sg = smem;                        // 128 x OSTR tile (34,816 B)
#pragma unroll
    for (int j = 0; j < 4; ++j) {
        int o = ohalf * 64 + j * 16 + row;
#pragma unroll
        for (int r = 0; r < 8; ++r) {
            int w = msub * 16 + khi * 8 + r;
            Sg[o * OSTR + w] = acc[j][r];
        }
    }
    __syncthreads();

    for (int e = tid; e < OO * 16; e += 256) {   // 128 rows x 16 float4
        int o  = e >> 4;
        int w4 = e & 15;
        float4 v = *(const float4*)&Sg[o * OSTR + w4 * 4];
        *(float4*)&out[(((size_t)b * OO + o) * HH + h) * WW + w4 * 4] = v;
    }
}

extern "C" void kernel_launch(void* const* d_in, const int* in_sizes, int n_in,
                              void* d_out, int out_size, void* d_ws, size_t ws_size,
                              hipStream_t stream) {
    const float* x       = (const float*)d_in[0];
    const float* shift_w = (const float*)d_in[1];
    const float* shift_b = (const float*)d_in[2];
    const float* mod_w   = (const float*)d_in[3];
    const float* mod_b   = (const float*)d_in[4];
    const float* conv_w  = (const float*)d_in[5];
    float* out = (float*)d_out;

    float* ws  = (float*)d_ws;
    float* xp  = ws;                                        // 8*66*66*64
    float* wtb = xp + (size_t)BB * HP * HP * CCH;           // 576*128 (paired rows)
    float* pmb = wtb + (size_t)KKT * OO;                    // 8*64*64*27

    int nxp = BB * HP * HP * CCH;
    pad_nhwc_kernel<<<(nxp + 255) / 256, 256, 0, stream>>>(x, xp);
    wtr_kernel<<<(KKT * OO + 255) / 256, 256, 0, stream>>>(conv_w, wtb);
    offmod_kernel<<<(BB * HH * WW) / 8, 256, 0, stream>>>(xp, shift_w, shift_b,
                                                          mod_w, mod_b, pmb);
    deform_gemm_kernel<<<BB * HH, 256, 0, stream>>>(xp, wtb, pmb, out);
}